// ROSA_1bit_LAYER_RODEO_58368605553264
// MI455X (gfx1250) — compile-verified
//
#include <hip/hip_runtime.h>
#include <stdint.h>

// Problem shape (from reference setup_inputs): x [B,T,C] fp32.
constexpr int B = 4, T = 2048, C = 128;
constexpr int W = T / 32;         // packed 32-bit words per sequence (64)
constexpr int NSEQ = B * C;       // 512 independent sequences
constexpr int MAXST = 2 * T;      // SAM needs <= 2T-1 states
constexpr int SPW = 4;            // sequences per workgroup (4 x 40KB = 160KB LDS)

// ---------------------------------------------------------------------------
// Kernel 1: binarize + bit-pack. Stages a 32(t) x 128(c) fp32 tile into LDS
// with CDNA5 async global->LDS loads, then packs sign bits: one u32 per
// channel per 32-step chunk, layout bits[b][w][c] (coalesced store).
// ---------------------------------------------------------------------------
__global__ void rosa_pack_kernel(const float* __restrict__ x,
                                 uint32_t* __restrict__ bits) {
  __shared__ float tile[32 * C];          // 16 KB
  const int b     = blockIdx.x >> 6;      // T/32 = 64 chunks per batch
  const int chunk = blockIdx.x & 63;
  const int tid   = threadIdx.x;          // 0..127 = channel
  const float* src = x + ((size_t)(b * T + chunk * 32)) * C;

#pragma unroll
  for (int r = 0; r < 32; ++r) {
    uint32_t voff = (uint32_t)((r * C + tid) * 4);
    // Low 32 bits of a flat shared-memory address are the LDS byte offset.
    uint32_t ldst = (uint32_t)(uintptr_t)(&tile[r * C + tid]);
    asm volatile("global_load_async_to_lds_b32 %0, %1, %2"
                 :: "v"(ldst), "v"(voff), "s"(src) : "memory");
  }
  asm volatile("s_wait_asynccnt 0" ::: "memory");
  __syncthreads();

  uint32_t wbits = 0u;
#pragma unroll
  for (int r = 0; r < 32; ++r)
    wbits |= (tile[r * C + tid] > 0.0f ? 1u : 0u) << r;
  bits[((size_t)b * W + chunk) * C + tid] = wbits;
}

// ---------------------------------------------------------------------------
// Kernel 2: online suffix automaton, one sequence per active lane, with the
// ENTIRE automaton resident in CDNA5's 320KB workgroup LDS (161KB used):
//   trans (2x16b packed) 64KB | len 32KB | link 32KB | lastpos 32KB | bits 1KB
// Every dependent hop on the serial critical path is an LDS access instead of
// an L2 access (~10x lower latency). 128 one-wave workgroups, 4 lanes active.
// ---------------------------------------------------------------------------
__global__ void rosa_sam_lds_kernel(const uint32_t* __restrict__ bitsG,
                                    const float* __restrict__ emb0,
                                    const float* __restrict__ emb1,
                                    float* __restrict__ out) {
  __shared__ uint32_t trS [MAXST * SPW];  // 64 KB  both transitions packed
  __shared__ uint16_t lenS[MAXST * SPW];  // 32 KB
  __shared__ int16_t  lnkS[MAXST * SPW];  // 32 KB  (-1 = no link)
  __shared__ uint16_t lstS[MAXST * SPW];  // 32 KB  last endpos per state
  __shared__ uint32_t blS [W * SPW];      //  1 KB  packed input bits

  const int lane  = threadIdx.x;          // 0..31
  const int b     = blockIdx.x >> 5;      // 32 blocks per batch
  const int cbase = (blockIdx.x & 31) * SPW;

  // Cooperative staging of the 4 sequences' packed bits (256 words).
  for (int i = lane; i < W * SPW; i += 32) {
    const int w = i >> 2, s = i & (SPW - 1);
    blS[w * SPW + s] = bitsG[((size_t)b * W + w) * C + (cbase + s)];
  }
  __syncthreads();
  if (lane >= SPW) return;                // 4 serial chains per wave

  const int c = cbase + lane;
  const float e0 = emb0[c];
  const float e1 = emb1[c];

#define LEN(s) lenS[(s) * SPW + lane]
#define LNK(s) lnkS[(s) * SPW + lane]
#define LST(s) lstS[(s) * SPW + lane]
#define TRN(s) trS [(s) * SPW + lane]
#define BIT(t) ((blS[((t) >> 5) * SPW + lane] >> ((t) & 31)) & 1u)

  // Root state.
  LEN(0) = 0; LNK(0) = -1; LST(0) = 0; TRN(0) = 0u;
  int nst = 1, last = 0, lenLast = 0;
  int cur = 0, l = 0;                     // matching-statistics state

  for (int t = 0; t < T; ++t) {
    const uint32_t cb = BIT(t);

    // Re-canonicalize (cur,l): clone splits may have re-homed the string.
    int lk = (int)LNK(cur);
    while (lk >= 0 && (int)LEN(lk) >= l) { cur = lk; lk = (int)LNK(cur); }

    // Matching-statistics step against the SAM of x[0..t-1].
    int p = cur;
    uint32_t tw = TRN(p);
    uint32_t nx = cb ? (tw >> 16) : (tw & 0xffffu);
    while (nx == 0u) {
      p = (int)LNK(p);
      if (p < 0) break;
      tw = TRN(p);
      nx = cb ? (tw >> 16) : (tw & 0xffffu);
    }
    int y;
    if (p < 0) { cur = 0; l = 0; y = -1; }
    else {
      const int lp = (int)LEN(p);
      l = (l < lp ? l : lp) + 1;
      cur = (int)nx;
      const int e = (int)LST(cur);        // most recent earlier end position
      y = (int)BIT(e + 1);
    }
    out[((size_t)(b * T + t)) * C + c] = (y < 0) ? 0.0f : (y ? e1 : e0);

    // sa_extend(cb) at position t.
    const int ncur = nst++;
    LEN(ncur) = (uint16_t)(lenLast + 1);
    TRN(ncur) = 0u;
    LST(ncur) = (uint16_t)t;
    int pp = last;
    while (pp >= 0) {
      uint32_t w2 = TRN(pp);
      uint32_t ex = cb ? (w2 >> 16) : (w2 & 0xffffu);
      if (ex != 0u) break;
      TRN(pp) = cb ? (w2 | ((uint32_t)ncur << 16)) : (w2 | (uint32_t)ncur);
      pp = (int)LNK(pp);
    }
    if (pp < 0) {
      LNK(ncur) = 0;
    } else {
      const uint32_t w2 = TRN(pp);
      const int q   = (int)(cb ? (w2 >> 16) : (w2 & 0xffffu));
      const int lpp = (int)LEN(pp);
      if (lpp + 1 == (int)LEN(q)) {
        LNK(ncur) = (int16_t)q;
      } else {
        const int cl = nst++;
        LEN(cl) = (uint16_t)(lpp + 1);
        LNK(cl) = LNK(q);
        TRN(cl) = TRN(q);
        LST(cl) = LST(q);                 // clone shares q's endpos history
        while (pp >= 0) {
          uint32_t w3 = TRN(pp);
          int ex = (int)(cb ? (w3 >> 16) : (w3 & 0xffffu));
          if (ex != q) break;
          TRN(pp) = cb ? ((w3 & 0x0000ffffu) | ((uint32_t)cl << 16))
                       : ((w3 & 0xffff0000u) | (uint32_t)cl);
          pp = (int)LNK(pp);
        }
        LNK(q)    = (int16_t)cl;
        LNK(ncur) = (int16_t)cl;
      }
    }
    last = ncur;
    ++lenLast;
    // Position t is a new endpos for every state on the suffix-link chain.
    for (int s = last; s >= 0; s = (int)LNK(s)) LST(s) = (uint16_t)t;
  }
#undef LEN
#undef LNK
#undef LST
#undef TRN
#undef BIT
}

extern "C" void kernel_launch(void* const* d_in, const int* in_sizes, int n_in,
                              void* d_out, int out_size, void* d_ws, size_t ws_size,
                              hipStream_t stream) {
  (void)in_sizes; (void)n_in; (void)out_size; (void)ws_size;
  const float* x    = (const float*)d_in[0];
  const float* emb0 = (const float*)d_in[1];
  const float* emb1 = (const float*)d_in[2];
  float* out = (float*)d_out;

  uint32_t* bits = (uint32_t*)d_ws;       // only 128 KB of workspace needed

  rosa_pack_kernel<<<dim3(B * (T / 32)), dim3(C), 0, stream>>>(x, bits);
  rosa_sam_lds_kernel<<<dim3(NSEQ / SPW), dim3(32), 0, stream>>>(
      bits, emb0, emb1, out);
}